// GAT_12292196401221
// MI455X (gfx1250) — compile-verified
//
#include <hip/hip_runtime.h>
#include <hip/hip_bf16.h>
#include <math.h>

#define F1 128          // input features / layer-1 width (H*C = 8*16)
#define H1 8            // heads layer 1
#define C1 16           // channels/head layer 1
#define NC 32           // classes (layer-2 width)
#define NEG_SLOPE 0.2f

typedef __attribute__((ext_vector_type(2))) float v2f;
typedef __attribute__((ext_vector_type(8))) float v8f;

// ---------------- utility ----------------

__device__ __forceinline__ float atomicMaxF(float* addr, float val) {
    unsigned int* ua = (unsigned int*)addr;
    unsigned int old = __hip_atomic_load(ua, __ATOMIC_RELAXED, __HIP_MEMORY_SCOPE_AGENT);
    while (__uint_as_float(old) < val) {
        unsigned int assumed = old;
        old = atomicCAS(ua, assumed, __float_as_uint(val));
        if (old == assumed) break;
    }
    return __uint_as_float(old);
}

__global__ void fill_kernel(float* __restrict__ p, float v, int n) {
    int i = blockIdx.x * blockDim.x + threadIdx.x;
    if (i < n) p[i] = v;
}

// d_out[n*32+c] = b2[c]  (final aggregation atomically adds on top)
__global__ void init_out_kernel(float* __restrict__ out, const float* __restrict__ b2, int n) {
    int i = blockIdx.x * blockDim.x + threadIdx.x;
    if (i < n) out[i] = b2[i & (NC - 1)];
}

// ---------------- WMMA GEMMs (fp32 matrix core path) ----------------
// Layer 1: H = X[N,128] @ W1[128,128].  One wave per 16x16 D-tile,
// 8 waves/block -> one block covers a 16-row stripe x all 128 cols.
__global__ void gemm1_wmma(const float* __restrict__ X, const float* __restrict__ W,
                           float* __restrict__ Hout, int N) {
    const int lane   = threadIdx.x & 31;
    const int wave   = threadIdx.x >> 5;           // column tile 0..7
    const int rowB   = blockIdx.x * 16;
    const int colB   = wave * 16;
    const int r15    = lane & 15;
    const int khalf  = (lane >> 4) * 2;            // K sub-offset: 0 or 2

    const int aRow   = rowB + r15;
    const float* xr  = X + (size_t)((aRow < N) ? aRow : 0) * F1;
    const int bCol   = colB + r15;

    v8f acc = {};
    #pragma unroll
    for (int k0 = 0; k0 < F1; k0 += 4) {
        v2f a, b;
        a.x = xr[k0 + khalf];
        a.y = xr[k0 + khalf + 1];
        b.x = W[(size_t)(k0 + khalf)     * F1 + bCol];
        b.y = W[(size_t)(k0 + khalf + 1) * F1 + bCol];
        acc = __builtin_amdgcn_wmma_f32_16x16x4_f32(false, a, false, b,
                                                    (short)0, acc, false, false);
    }
    const int rOff = (lane >> 4) * 8;
    #pragma unroll
    for (int r = 0; r < 8; ++r) {
        int row = rowB + rOff + r;
        if (row < N) Hout[(size_t)row * F1 + colB + r15] = acc[r];
    }
}

// Layer 2: G = H2[N,128] @ W2[128,32].  2 waves/block (two 16-col tiles).
__global__ void gemm2_wmma(const float* __restrict__ X, const float* __restrict__ W,
                           float* __restrict__ G, int N) {
    const int lane  = threadIdx.x & 31;
    const int wave  = threadIdx.x >> 5;            // column tile 0..1
    const int rowB  = blockIdx.x * 16;
    const int colB  = wave * 16;
    const int r15   = lane & 15;
    const int khalf = (lane >> 4) * 2;

    const int aRow  = rowB + r15;
    const float* xr = X + (size_t)((aRow < N) ? aRow : 0) * F1;
    const int bCol  = colB + r15;

    v8f acc = {};
    #pragma unroll
    for (int k0 = 0; k0 < F1; k0 += 4) {
        v2f a, b;
        a.x = xr[k0 + khalf];
        a.y = xr[k0 + khalf + 1];
        b.x = W[(size_t)(k0 + khalf)     * NC + bCol];
        b.y = W[(size_t)(k0 + khalf + 1) * NC + bCol];
        acc = __builtin_amdgcn_wmma_f32_16x16x4_f32(false, a, false, b,
                                                    (short)0, acc, false, false);
    }
    const int rOff = (lane >> 4) * 8;
    #pragma unroll
    for (int r = 0; r < 8; ++r) {
        int row = rowB + rOff + r;
        if (row < N) G[(size_t)row * NC + colB + r15] = acc[r];
    }
}

// ---------------- per-node attention coefficients ----------------
// a_src[n,h] = sum_c h[n,h,c]*att_src[h,c]   (layer 1: H=8, C=16)
__global__ void attn_coef1(const float* __restrict__ Hm,
                           const float* __restrict__ att_s, const float* __restrict__ att_d,
                           float* __restrict__ as, float* __restrict__ ad, int N) {
    int i = blockIdx.x * blockDim.x + threadIdx.x;
    if (i >= N * H1) return;
    int n = i >> 3, h = i & 7;
    const float* hp = Hm + (size_t)n * F1 + h * C1;
    float ss = 0.f, sd = 0.f;
    #pragma unroll
    for (int c = 0; c < C1; ++c) {
        float v = hp[c];
        ss += v * att_s[h * C1 + c];
        sd += v * att_d[h * C1 + c];
    }
    as[i] = ss; ad[i] = sd;
}

// layer 2: H=1, C=32
__global__ void attn_coef2(const float* __restrict__ G,
                           const float* __restrict__ att_s, const float* __restrict__ att_d,
                           float* __restrict__ as, float* __restrict__ ad, int N) {
    int n = blockIdx.x * blockDim.x + threadIdx.x;
    if (n >= N) return;
    const float* g = G + (size_t)n * NC;
    float ss = 0.f, sd = 0.f;
    #pragma unroll
    for (int c = 0; c < NC; ++c) { ss += g[c] * att_s[c]; sd += g[c] * att_d[c]; }
    as[n] = ss; ad[n] = sd;
}

// ---------------- edge passes (segment softmax) ----------------
template <int HH>
__global__ void edge_max_kernel(const int* __restrict__ src, const int* __restrict__ dst,
                                const float* __restrict__ as, const float* __restrict__ ad,
                                float* __restrict__ amax, int E, int N) {
    int e = blockIdx.x * blockDim.x + threadIdx.x;
    if (e >= E + N) return;
    int s = (e < E) ? src[e] : (e - E);
    int d = (e < E) ? dst[e] : (e - E);
    s = min(max(s, 0), N - 1); d = min(max(d, 0), N - 1);
    #pragma unroll
    for (int h = 0; h < HH; ++h) {
        float a = as[s * HH + h] + ad[d * HH + h];
        a = (a > 0.f) ? a : NEG_SLOPE * a;
        atomicMaxF(&amax[d * HH + h], a);
    }
}

template <int HH>
__global__ void edge_sum_kernel(const int* __restrict__ src, const int* __restrict__ dst,
                                const float* __restrict__ as, const float* __restrict__ ad,
                                const float* __restrict__ amax, float* __restrict__ denom,
                                int E, int N) {
    int e = blockIdx.x * blockDim.x + threadIdx.x;
    if (e >= E + N) return;
    int s = (e < E) ? src[e] : (e - E);
    int d = (e < E) ? dst[e] : (e - E);
    s = min(max(s, 0), N - 1); d = min(max(d, 0), N - 1);
    #pragma unroll
    for (int h = 0; h < HH; ++h) {
        float a = as[s * HH + h] + ad[d * HH + h];
        a = (a > 0.f) ? a : NEG_SLOPE * a;
        atomicAdd(&denom[d * HH + h], expf(a - amax[d * HH + h]));
    }
}

// layer-1 aggregation: one wave per edge; lane owns 4 channels (128/32),
// head = lane/4 (4 channels never straddle a head since C1=16).
__global__ void edge_agg1(const int* __restrict__ src, const int* __restrict__ dst,
                          const float* __restrict__ Hm,
                          const float* __restrict__ as, const float* __restrict__ ad,
                          const float* __restrict__ amax, const float* __restrict__ denom,
                          float* __restrict__ out, int E, int N) {
    int lane = threadIdx.x & 31;
    int e = (blockIdx.x * blockDim.x + threadIdx.x) >> 5;
    if (e >= E + N) return;
    int s = (e < E) ? src[e] : (e - E);
    int d = (e < E) ? dst[e] : (e - E);
    s = min(max(s, 0), N - 1); d = min(max(d, 0), N - 1);
    int h = lane >> 2;
    float a = as[s * H1 + h] + ad[d * H1 + h];
    a = (a > 0.f) ? a : NEG_SLOPE * a;
    float w = expf(a - amax[d * H1 + h]) / (denom[d * H1 + h] + 1e-16f);
    const float4 v = *(const float4*)(Hm + (size_t)s * F1 + lane * 4);
    float* o = out + (size_t)d * F1 + lane * 4;
    atomicAdd(o + 0, v.x * w); atomicAdd(o + 1, v.y * w);
    atomicAdd(o + 2, v.z * w); atomicAdd(o + 3, v.w * w);
}

// layer-2 aggregation: one wave per edge; lane owns 1 of 32 classes. Adds into d_out (pre-biased).
__global__ void edge_agg2(const int* __restrict__ src, const int* __restrict__ dst,
                          const float* __restrict__ G,
                          const float* __restrict__ as, const float* __restrict__ ad,
                          const float* __restrict__ amax, const float* __restrict__ denom,
                          float* __restrict__ out, int E, int N) {
    int lane = threadIdx.x & 31;
    int e = (blockIdx.x * blockDim.x + threadIdx.x) >> 5;
    if (e >= E + N) return;
    int s = (e < E) ? src[e] : (e - E);
    int d = (e < E) ? dst[e] : (e - E);
    s = min(max(s, 0), N - 1); d = min(max(d, 0), N - 1);
    float a = as[s] + ad[d];
    a = (a > 0.f) ? a : NEG_SLOPE * a;
    float w = expf(a - amax[d]) / (denom[d] + 1e-16f);
    atomicAdd(out + (size_t)d * NC + lane, G[(size_t)s * NC + lane] * w);
}

// h2 = elu(out1 + b1), in place
__global__ void bias_elu_kernel(float* __restrict__ p, const float* __restrict__ b, int n) {
    int i = blockIdx.x * blockDim.x + threadIdx.x;
    if (i >= n) return;
    float v = p[i] + b[i & (F1 - 1)];
    p[i] = (v > 0.f) ? v : (expf(v) - 1.f);
}

// ---------------- host launch ----------------
extern "C" void kernel_launch(void* const* d_in, const int* in_sizes, int n_in,
                              void* d_out, int out_size, void* d_ws, size_t ws_size,
                              hipStream_t stream) {
    const float* x    = (const float*)d_in[0];
    const int*   ei   = (const int*)  d_in[1];
    const float* W1   = (const float*)d_in[2];
    const float* ats1 = (const float*)d_in[3];
    const float* atd1 = (const float*)d_in[4];
    const float* b1   = (const float*)d_in[5];
    const float* W2   = (const float*)d_in[6];
    const float* ats2 = (const float*)d_in[7];
    const float* atd2 = (const float*)d_in[8];
    const float* b2   = (const float*)d_in[9];
    float* out = (float*)d_out;

    const int N = in_sizes[0] / F1;
    const int E = in_sizes[1] / 2;
    const int Etot = E + N;
    const int* src = ei;
    const int* dst = ei + E;

    // workspace layout (floats)
    float* ws     = (float*)d_ws;
    float* h1     = ws;                      // N*128
    float* as1    = h1   + (size_t)N * F1;   // N*8
    float* ad1    = as1  + (size_t)N * H1;
    float* amax1  = ad1  + (size_t)N * H1;
    float* den1   = amax1+ (size_t)N * H1;
    float* out1   = den1 + (size_t)N * H1;   // N*128 (becomes h2 after ELU)
    float* g2     = out1 + (size_t)N * F1;   // N*32
    float* as2    = g2   + (size_t)N * NC;   // N
    float* ad2    = as2  + (size_t)N;
    float* amax2  = ad2  + (size_t)N;
    float* den2   = amax2+ (size_t)N;

    const int B = 256;
    const int rowTiles = (N + 15) / 16;
    const int edgeBlk  = (Etot + B - 1) / B;
    const int waveBlk  = (Etot + (B / 32) - 1) / (B / 32);   // wave-per-edge grids

    // ---- init accumulators (graph-replay safe) ----
    fill_kernel<<<(N*H1 + B-1)/B, B, 0, stream>>>(amax1, -INFINITY, N*H1);
    fill_kernel<<<(N*H1 + B-1)/B, B, 0, stream>>>(den1,  0.f,       N*H1);
    fill_kernel<<<(N*F1 + B-1)/B, B, 0, stream>>>(out1,  0.f,       N*F1);
    fill_kernel<<<(N    + B-1)/B, B, 0, stream>>>(amax2, -INFINITY, N);
    fill_kernel<<<(N    + B-1)/B, B, 0, stream>>>(den2,  0.f,       N);
    init_out_kernel<<<(N*NC + B-1)/B, B, 0, stream>>>(out, b2, N*NC);

    // ---- layer 1 ----
    gemm1_wmma<<<rowTiles, 256, 0, stream>>>(x, W1, h1, N);
    attn_coef1<<<(N*H1 + B-1)/B, B, 0, stream>>>(h1, ats1, atd1, as1, ad1, N);
    edge_max_kernel<H1><<<edgeBlk, B, 0, stream>>>(src, dst, as1, ad1, amax1, E, N);
    edge_sum_kernel<H1><<<edgeBlk, B, 0, stream>>>(src, dst, as1, ad1, amax1, den1, E, N);
    edge_agg1<<<waveBlk, B, 0, stream>>>(src, dst, h1, as1, ad1, amax1, den1, out1, E, N);
    bias_elu_kernel<<<(N*F1 + B-1)/B, B, 0, stream>>>(out1, b1, N*F1);

    // ---- layer 2 ----
    gemm2_wmma<<<rowTiles, 64, 0, stream>>>(out1, W2, g2, N);
    attn_coef2<<<(N + B-1)/B, B, 0, stream>>>(g2, ats2, atd2, as2, ad2, N);
    edge_max_kernel<1><<<edgeBlk, B, 0, stream>>>(src, dst, as2, ad2, amax2, E, N);
    edge_sum_kernel<1><<<edgeBlk, B, 0, stream>>>(src, dst, as2, ad2, amax2, den2, E, N);
    edge_agg2<<<waveBlk, B, 0, stream>>>(src, dst, g2, as2, ad2, amax2, den2, out, E, N);
    (void)n_in; (void)out_size; (void)ws_size;
}